// SelfAttention_90967407329792
// MI455X (gfx1250) — compile-verified
//
#include <hip/hip_runtime.h>
#include <hip/hip_bf16.h>

// ---------------------------------------------------------------------------
// Self-attention (B=8, N=1029, C=768, H=12, d=64) in fp32 using CDNA5
// V_WMMA_F32_16X16X4_F32 + GLOBAL_LOAD_ASYNC_TO_LDS_B128 staging.
// Flash-attention fused softmax: the 8*12*1029^2 score tensor never exists.
// ---------------------------------------------------------------------------

typedef __attribute__((ext_vector_type(2))) float v2f;
typedef __attribute__((ext_vector_type(8))) float v8f;

#define Bdim 8
#define Nseq 1029
#define Cdim 768
#define Hn   12
#define Dh   64
#define Rrot 1024
#define PREFIX (Nseq - Rrot)     /* 5 */
#define NT   ((Nseq + 15) / 16)  /* 65 q/kv tiles of 16 */
#define NT32 ((Nseq + 31) / 32)  /* 33 row tiles of 32 */
#define QGRP ((NT + 7) / 8)      /* 9 groups of 8 q-tiles */
#define KPITCH 68                /* LDS row pitch (floats): 64+4 banks-spread */

// D = A(16x4,f32) * B(4x16,f32) + C(16x16,f32)
// A frag (ISA 7.12.2): lanes 0-15 -> rows M with {K0,K1}; lanes 16-31 {K2,K3}.
// B frag symmetric: {B[k0][n], B[k0+1][n]}, n = lane&15, k0 = 2*(lane>>4).
// C/D: VGPR r holds row (r + 8*(lane>>4)), col = lane&15.
__device__ __forceinline__ v8f wmma_f32(v2f a, v2f b, v8f c) {
  return __builtin_amdgcn_wmma_f32_16x16x4_f32(false, a, false, b, (short)0, c,
                                               false, false);
}

// CDNA5 async global->LDS copy, 16B per lane, tracked with ASYNCcnt.
__device__ __forceinline__ void async_ld16(unsigned int lds_byte,
                                           unsigned long long gaddr) {
  asm volatile("global_load_async_to_lds_b128 %0, %1, off" ::"v"(lds_byte),
               "v"(gaddr)
               : "memory");
}
__device__ __forceinline__ void wait_async0() {
  asm volatile("s_wait_asynccnt 0x0" ::: "memory");
}
__device__ __forceinline__ void wait_ds0() {
  asm volatile("s_wait_dscnt 0x0" ::: "memory");
}

// ---------------------------------------------------------------------------
// Kernel 1: qkv = x @ w_qkv^T scattered to q/k/v in [B,H,N,64] layout.
// One wave computes a 32(row) x 64(col) tile: 2 A-frags x 4 B-frags per
// k-step feed 8 WMMAs (0.75 loads/WMMA).
// ---------------------------------------------------------------------------
__global__ __launch_bounds__(256) void qkv_gemm_kernel(
    const float* __restrict__ x, const float* __restrict__ w,
    float* __restrict__ qb, float* __restrict__ kb, float* __restrict__ vb) {
  const int wid  = (blockIdx.x << 3) | (threadIdx.x >> 5);
  const int lane = threadIdx.x & 31;
  const int l16  = lane & 15;
  const int half = lane >> 4;

  const int cg = wid % 36;              // 36 groups of 64 output cols (3C)
  const int nt = (wid / 36) % NT32;     // 33 row tiles of 32
  const int b  = wid / (36 * NT32);

  const int r0 = nt * 32 + l16;
  const int r1 = r0 + 16;
  const float* xr0 = x + ((size_t)b * Nseq + (r0 < Nseq ? r0 : Nseq - 1)) * Cdim;
  const float* xr1 = x + ((size_t)b * Nseq + (r1 < Nseq ? r1 : Nseq - 1)) * Cdim;
  const float* w0  = w + ((size_t)cg * 64 + l16) * Cdim;

  v8f acc[2][4] = {};
#pragma unroll 4
  for (int t = 0; t < Cdim / 4; ++t) {
    const int k0 = 4 * t + 2 * half;
    v2f a0; a0.x = xr0[k0];               a0.y = xr0[k0 + 1];
    v2f a1; a1.x = xr1[k0];               a1.y = xr1[k0 + 1];
    v2f b0; b0.x = w0[k0];                b0.y = w0[k0 + 1];
    v2f b1; b1.x = w0[16 * Cdim + k0];    b1.y = w0[16 * Cdim + k0 + 1];
    v2f b2; b2.x = w0[32 * Cdim + k0];    b2.y = w0[32 * Cdim + k0 + 1];
    v2f b3; b3.x = w0[48 * Cdim + k0];    b3.y = w0[48 * Cdim + k0 + 1];
    acc[0][0] = wmma_f32(a0, b0, acc[0][0]);
    acc[1][0] = wmma_f32(a1, b0, acc[1][0]);
    acc[0][1] = wmma_f32(a0, b1, acc[0][1]);
    acc[1][1] = wmma_f32(a1, b1, acc[1][1]);
    acc[0][2] = wmma_f32(a0, b2, acc[0][2]);
    acc[1][2] = wmma_f32(a1, b2, acc[1][2]);
    acc[0][3] = wmma_f32(a0, b3, acc[0][3]);
    acc[1][3] = wmma_f32(a1, b3, acc[1][3]);
  }

  const int which = cg / Hn;  // 0=q 1=k 2=v
  const int h     = cg % Hn;
  float* dst = (which == 0 ? qb : which == 1 ? kb : vb) +
               (size_t)(b * Hn + h) * Nseq * Dh;
#pragma unroll
  for (int i = 0; i < 2; ++i) {
#pragma unroll
    for (int j = 0; j < 4; ++j) {
#pragma unroll
      for (int r = 0; r < 8; ++r) {
        const int n = nt * 32 + i * 16 + r + 8 * half;
        if (n < Nseq) dst[(size_t)n * Dh + j * 16 + l16] = acc[i][j][r];
      }
    }
  }
}

// ---------------------------------------------------------------------------
// Kernel 2: in-place RoPE on q and k for n >= PREFIX; each thread owns a
// (dd, dd+32) pair so the rotation has no read/write race.
// ---------------------------------------------------------------------------
__global__ __launch_bounds__(256) void rope_kernel(
    float* __restrict__ qb, float* __restrict__ kb,
    const float* __restrict__ sn, const float* __restrict__ cs) {
  int idx = blockIdx.x * 256 + threadIdx.x;  // (b,h,r,dd2), dd2 in [0,32)
  const int dd = idx & 31;   idx >>= 5;
  const int r  = idx & 1023; idx >>= 10;
  const int h  = idx % Hn;
  const int b  = idx / Hn;
  const int n  = r + PREFIX;
  const size_t base = ((size_t)(b * Hn + h) * Nseq + n) * Dh;
  const size_t sb   = ((size_t)b * Rrot + r) * Dh;
  const float s1 = sn[sb + dd],      c1 = cs[sb + dd];
  const float s2 = sn[sb + dd + 32], c2 = cs[sb + dd + 32];
  const float q1 = qb[base + dd], q2 = qb[base + dd + 32];
  qb[base + dd]      = q1 * c1 - q2 * s1;
  qb[base + dd + 32] = q2 * c2 + q1 * s2;
  const float k1 = kb[base + dd], k2 = kb[base + dd + 32];
  kb[base + dd]      = k1 * c1 - k2 * s1;
  kb[base + dd + 32] = k2 * c2 + k1 * s2;
}

// ---------------------------------------------------------------------------
// Kernel 3: flash attention. Block = 8 waves = 8 consecutive q-tiles of one
// (b,h). Each KV tile (16x64 K and V) is staged once per block into LDS with
// global_load_async_to_lds_b128 and consumed by all 8 waves (8x less global
// traffic). Per wave per KV tile: 16 WMMAs for S = QK^T, online softmax with
// 16-lane-half reductions, LDS C->A transpose of P, 16 WMMAs for O += P*V.
// ---------------------------------------------------------------------------
__global__ __launch_bounds__(256) void attn_kernel(
    const float* __restrict__ qb, const float* __restrict__ kb,
    const float* __restrict__ vb, float* __restrict__ ob) {
  __shared__ float Ktile[16 * KPITCH];
  __shared__ float Vtile[16 * KPITCH];
  __shared__ float stage[8][16 * 17];  // per-wave P transpose buffer

  const int tid   = threadIdx.x;
  const int wslot = tid >> 5;
  const int lane  = tid & 31;
  const int l16   = lane & 15;
  const int half  = lane >> 4;

  const int gq = blockIdx.x % QGRP;
  const int h  = (blockIdx.x / QGRP) % Hn;
  const int b  = blockIdx.x / (QGRP * Hn);
  const int qt = gq * 8 + wslot;
  const bool active = qt < NT;

  const size_t hb = (size_t)(b * Hn + h) * Nseq * Dh;
  const float* Q = qb + hb;
  const float* K = kb + hb;
  const float* V = vb + hb;
  float* st = stage[wslot];

  // cooperative-load mapping: thread -> (row 0..15, 16B chunk 0..15)
  const int crow = tid >> 4;
  const int ccol = tid & 15;
  const unsigned int klds =
      (unsigned int)(size_t)(void*)(Ktile + crow * KPITCH + ccol * 4);
  const unsigned int vlds =
      (unsigned int)(size_t)(void*)(Vtile + crow * KPITCH + ccol * 4);

  // Q fragments over d=64 (16 k-steps), pre-scaled by d^-0.5 = 0.125
  const int qrow = qt * 16 + l16;
  const float* qr = Q + (size_t)(qrow < Nseq ? qrow : Nseq - 1) * Dh;
  v2f qf[16];
#pragma unroll
  for (int t = 0; t < 16; ++t) {
    const int k0 = 4 * t + 2 * half;
    qf[t].x = qr[k0] * 0.125f;
    qf[t].y = qr[k0 + 1] * 0.125f;
  }

  v8f o0 = {}, o1 = {}, o2 = {}, o3 = {};
  float mrow[8], lrow[8];
#pragma unroll
  for (int r = 0; r < 8; ++r) { mrow[r] = -3.0e38f; lrow[r] = 0.0f; }

  for (int kv = 0; kv < NT; ++kv) {
    const int kvbase = kv * 16;
    // ---- async stage K and V tiles into LDS (all 256 threads) ----
    {
      const int grow = kvbase + crow;
      const size_t gro = (size_t)(grow < Nseq ? grow : Nseq - 1) * Dh + ccol * 4;
      async_ld16(klds, (unsigned long long)(const void*)(K + gro));
      async_ld16(vlds, (unsigned long long)(const void*)(V + gro));
      wait_async0();
    }
    __syncthreads();  // all waves' async copies complete -> LDS valid

    if (active) {
      // ---- S = (Q/sqrt(d)) K^T ; B frag from LDS, 64-bank-clean pitch ----
      const float* kl = Ktile + l16 * KPITCH;
      v8f s = {};
#pragma unroll
      for (int t = 0; t < 16; ++t) {
        const int k0 = 4 * t + 2 * half;
        v2f bb; bb.x = kl[k0]; bb.y = kl[k0 + 1];
        s = wmma_f32(qf[t], bb, s);
      }
      if (kvbase + l16 >= Nseq) {  // mask padded kv columns (finite sentinel)
#pragma unroll
        for (int r = 0; r < 8; ++r) s[r] = -3.0e38f;
      }
      // ---- online softmax (rows live across 16-lane halves) ----
      float pr[8];
#pragma unroll
      for (int r = 0; r < 8; ++r) {
        float mx = s[r];
        mx = fmaxf(mx, __shfl_xor(mx, 1));
        mx = fmaxf(mx, __shfl_xor(mx, 2));
        mx = fmaxf(mx, __shfl_xor(mx, 4));
        mx = fmaxf(mx, __shfl_xor(mx, 8));
        const float mnew = fmaxf(mrow[r], mx);
        const float corr = __expf(mrow[r] - mnew);
        const float p    = __expf(s[r] - mnew);
        float psum = p;
        psum += __shfl_xor(psum, 1);
        psum += __shfl_xor(psum, 2);
        psum += __shfl_xor(psum, 4);
        psum += __shfl_xor(psum, 8);
        lrow[r] = lrow[r] * corr + psum;
        mrow[r] = mnew;
        pr[r]   = p;
        o0[r] *= corr; o1[r] *= corr; o2[r] *= corr; o3[r] *= corr;
      }
      // ---- stage P: C-layout -> [row][col] (per-wave buffer) ----
#pragma unroll
      for (int r = 0; r < 8; ++r) st[(r + 8 * half) * 17 + l16] = pr[r];
      wait_ds0();
      // ---- O += P * V (V from LDS) ----
#pragma unroll
      for (int t = 0; t < 4; ++t) {
        const int k0 = 4 * t + 2 * half;
        v2f a;
        a.x = st[l16 * 17 + k0];
        a.y = st[l16 * 17 + k0 + 1];
        const float* v0 = Vtile + k0 * KPITCH;
        const float* v1 = Vtile + (k0 + 1) * KPITCH;
        v2f b0; b0.x = v0[l16];      b0.y = v1[l16];
        v2f b1; b1.x = v0[16 + l16]; b1.y = v1[16 + l16];
        v2f b2; b2.x = v0[32 + l16]; b2.y = v1[32 + l16];
        v2f b3; b3.x = v0[48 + l16]; b3.y = v1[48 + l16];
        o0 = wmma_f32(a, b0, o0);
        o1 = wmma_f32(a, b1, o1);
        o2 = wmma_f32(a, b2, o2);
        o3 = wmma_f32(a, b3, o3);
      }
      wait_ds0();  // WAR: P stage + K/V reads done before next overwrite
    }
    __syncthreads();  // protect K/V tiles until every wave consumed them
  }

  if (active) {
    v8f oacc[4] = {o0, o1, o2, o3};
#pragma unroll
    for (int j = 0; j < 4; ++j) {
#pragma unroll
      for (int r = 0; r < 8; ++r) {
        const int n = qt * 16 + r + 8 * half;
        if (n < Nseq) {
          const float inv = 1.0f / lrow[r];
          ob[((size_t)b * Nseq + n) * Cdim + h * Dh + j * 16 + l16] =
              oacc[j][r] * inv;
        }
      }
    }
  }
}

// ---------------------------------------------------------------------------
// Kernel 4: out = o @ w_proj^T + b_proj.  Same 32x64-per-wave skeleton.
// ---------------------------------------------------------------------------
__global__ __launch_bounds__(256) void proj_kernel(
    const float* __restrict__ ob, const float* __restrict__ w,
    const float* __restrict__ bias, float* __restrict__ out) {
  const int wid  = (blockIdx.x << 3) | (threadIdx.x >> 5);
  const int lane = threadIdx.x & 31;
  const int l16  = lane & 15;
  const int half = lane >> 4;

  const int cg = wid % 12;
  const int nt = (wid / 12) % NT32;
  const int b  = wid / (12 * NT32);

  const int r0 = nt * 32 + l16;
  const int r1 = r0 + 16;
  const float* xr0 = ob + ((size_t)b * Nseq + (r0 < Nseq ? r0 : Nseq - 1)) * Cdim;
  const float* xr1 = ob + ((size_t)b * Nseq + (r1 < Nseq ? r1 : Nseq - 1)) * Cdim;
  const float* w0  = w + ((size_t)cg * 64 + l16) * Cdim;

  v8f acc[2][4] = {};
#pragma unroll 4
  for (int t = 0; t < Cdim / 4; ++t) {
    const int k0 = 4 * t + 2 * half;
    v2f a0; a0.x = xr0[k0];               a0.y = xr0[k0 + 1];
    v2f a1; a1.x = xr1[k0];               a1.y = xr1[k0 + 1];
    v2f b0; b0.x = w0[k0];                b0.y = w0[k0 + 1];
    v2f b1; b1.x = w0[16 * Cdim + k0];    b1.y = w0[16 * Cdim + k0 + 1];
    v2f b2; b2.x = w0[32 * Cdim + k0];    b2.y = w0[32 * Cdim + k0 + 1];
    v2f b3; b3.x = w0[48 * Cdim + k0];    b3.y = w0[48 * Cdim + k0 + 1];
    acc[0][0] = wmma_f32(a0, b0, acc[0][0]);
    acc[1][0] = wmma_f32(a1, b0, acc[1][0]);
    acc[0][1] = wmma_f32(a0, b1, acc[0][1]);
    acc[1][1] = wmma_f32(a1, b1, acc[1][1]);
    acc[0][2] = wmma_f32(a0, b2, acc[0][2]);
    acc[1][2] = wmma_f32(a1, b2, acc[1][2]);
    acc[0][3] = wmma_f32(a0, b3, acc[0][3]);
    acc[1][3] = wmma_f32(a1, b3, acc[1][3]);
  }

#pragma unroll
  for (int i = 0; i < 2; ++i) {
#pragma unroll
    for (int j = 0; j < 4; ++j) {
#pragma unroll
      for (int r = 0; r < 8; ++r) {
        const int n = nt * 32 + i * 16 + r + 8 * half;
        const int c = cg * 64 + j * 16 + l16;
        if (n < Nseq)
          out[((size_t)b * Nseq + n) * Cdim + c] = acc[i][j][r] + bias[c];
      }
    }
  }
}

// ---------------------------------------------------------------------------
extern "C" void kernel_launch(void* const* d_in, const int* in_sizes, int n_in,
                              void* d_out, int out_size, void* d_ws,
                              size_t ws_size, hipStream_t stream) {
  const float* x     = (const float*)d_in[0];
  const float* sn    = (const float*)d_in[1];
  const float* cs    = (const float*)d_in[2];
  const float* wqkv  = (const float*)d_in[3];
  const float* wproj = (const float*)d_in[4];
  const float* bproj = (const float*)d_in[5];
  float* out = (float*)d_out;

  const size_t per = (size_t)Bdim * Hn * Nseq * Dh;  // 6,322,176 floats
  float* qbuf = (float*)d_ws;
  float* kbuf = qbuf + per;
  float* vbuf = kbuf + per;
  float* obuf = vbuf + per;  // [B,N,C]

  // 1) QKV projection: 8*33*36 waves, 8 waves/block
  qkv_gemm_kernel<<<(Bdim * NT32 * 36) / 8, 256, 0, stream>>>(x, wqkv, qbuf,
                                                              kbuf, vbuf);
  // 2) RoPE in place on q,k
  rope_kernel<<<(Bdim * Hn * Rrot * 32) / 256, 256, 0, stream>>>(qbuf, kbuf, sn,
                                                                 cs);
  // 3) Flash attention: one block per (b,h,8-q-tile group)
  attn_kernel<<<Bdim * Hn * QGRP, 256, 0, stream>>>(qbuf, kbuf, vbuf, obuf);
  // 4) Output projection + bias: 8*33*12 waves
  proj_kernel<<<(Bdim * NT32 * 12) / 8, 256, 0, stream>>>(obuf, wproj, bproj,
                                                          out);
}